// Head_44796508897643
// MI455X (gfx1250) — compile-verified
//
#include <hip/hip_runtime.h>
#include <math.h>

#define B   256
#define N   8192
#define M   64
#define INW 256
#define SR  3
#define NOUT (M + SR + 3)   // 70
#define EPSF 1e-16f

typedef __attribute__((ext_vector_type(2))) float v2f;
typedef __attribute__((ext_vector_type(8))) float v8f;

__device__ __forceinline__ float softplusf(float x) {
    // numerically stable log(1+exp(x))
    return fmaxf(x, 0.0f) + log1pf(expf(-fabsf(x)));
}

// ---------------------------------------------------------------------------
// K1: outputs = ctrl @ fc_w^T + fc_b ; split into key/stren/gate/sharp/shift
// ---------------------------------------------------------------------------
__global__ __launch_bounds__(128) void ntm_fc_kernel(
        const float* __restrict__ ctrl,  const float* __restrict__ fc_w,
        const float* __restrict__ fc_b,
        float* __restrict__ key,   float* __restrict__ key_norm,
        float* __restrict__ stren, float* __restrict__ gate,
        float* __restrict__ sharp, float* __restrict__ shift) {
    const int b = blockIdx.x;
    const int j = threadIdx.x;
    __shared__ float outs[NOUT];
    __shared__ float kn[M];
    if (j < NOUT) {
        const float* c = ctrl + b * INW;
        const float* w = fc_w + j * INW;
        float acc = fc_b[j];
        #pragma unroll 8
        for (int i = 0; i < INW; ++i) acc = fmaf(c[i], w[i], acc);
        outs[j] = acc;
    }
    __syncthreads();
    if (j < M) {
        float k = tanhf(outs[j]);
        key[b * M + j] = k;
        kn[j] = k * k;
    }
    __syncthreads();
    if (j == 0) {
        float s = 0.f;
        for (int i = 0; i < M; ++i) s += kn[i];
        key_norm[b] = fmaxf(sqrtf(s), EPSF);
        stren[b] = softplusf(outs[M]);
        gate[b]  = 1.0f / (1.0f + expf(-outs[M + 1]));
        sharp[b] = 1.0f + softplusf(outs[M + 2]);
        float a0 = outs[M + 3], a1 = outs[M + 4], a2 = outs[M + 5];
        float mx = fmaxf(a0, fmaxf(a1, a2));
        float e0 = expf(a0 - mx), e1 = expf(a1 - mx), e2 = expf(a2 - mx);
        float inv = 1.0f / (e0 + e1 + e2);
        shift[b * SR + 0] = e0 * inv;
        shift[b * SR + 1] = e1 * inv;
        shift[b * SR + 2] = e2 * inv;
    }
}

// ---------------------------------------------------------------------------
// K2 (dominant, streams 512 MiB once): per wave, one (batch, 16-row) tile.
// A = memory rows (16x4 f32 chunks), B = key broadcast over all 16 columns,
// so D[i][j] == dot(row i) for every j.  Row L2-norms from the same data.
// ---------------------------------------------------------------------------
#define K2_WAVES 8
__global__ __launch_bounds__(256) void ntm_score_kernel(
        const float* __restrict__ memory, const float* __restrict__ key,
        const float* __restrict__ key_norm, const float* __restrict__ stren,
        float* __restrict__ score) {
    const int tid  = threadIdx.x;
    const int wave = tid >> 5;
    const int lane = tid & 31;
    const int half = lane >> 4;     // selects K pair within 4-wide chunk
    const int row  = lane & 15;     // A-matrix row (M index)
    const int b    = blockIdx.y;
    const int n0   = (blockIdx.x * K2_WAVES + wave) * 16;

    const float* mtile = memory + ((size_t)b * N + n0) * M;
    const float* keyb  = key + b * M;

    v8f acc = {};
    float nrm = 0.0f;
    #pragma unroll
    for (int k0 = 0; k0 < M; k0 += 4) {
        const int kk = k0 + half * 2;
        v2f a  = *(const v2f*)(mtile + (size_t)row * M + kk);  // A[row][kk..kk+1]
        v2f kb = *(const v2f*)(keyb + kk);                     // B[k][*] = key[k]
        nrm = fmaf(a.x, a.x, nrm);
        nrm = fmaf(a.y, a.y, nrm);
        acc = __builtin_amdgcn_wmma_f32_16x16x4_f32(
                  false, a, false, kb, (short)0, acc, false, false);
    }

    __shared__ float sdot[K2_WAVES][16];
    __shared__ float snrm[K2_WAVES][16];
    // D layout: VGPR r holds row r (lanes 0-15) / row r+8 (lanes 16-31);
    // all columns are equal, so lane 0 and lane 16 publish the 16 row dots.
    if (row == 0) {
        const int off = half * 8;
        #pragma unroll
        for (int i = 0; i < 8; ++i) sdot[wave][off + i] = acc[i];
    }
    // lane r covers K = {0,1} mod 4, lane r+16 covers K = {2,3} mod 4
    float n2 = nrm + __shfl_xor(nrm, 16, 32);
    if (lane < 16) snrm[wave][lane] = n2;
    __syncthreads();
    if (lane < 16) {
        float mn = fmaxf(sqrtf(snrm[wave][lane]), EPSF);
        float sc = stren[b] * sdot[wave][lane] / (mn * key_norm[b]);
        score[(size_t)b * N + n0 + lane] = sc;
    }
}

// ---------------------------------------------------------------------------
// K3: per-batch softmax over N, gate interpolation, circular 3-tap shift,
// sharpening; per-batch partial sum of w_tmp (deterministic tree reduce).
// score/w_tmp may alias (same buffer): reads finish in pass 1, writes in 4.
// ---------------------------------------------------------------------------
__global__ __launch_bounds__(256) void ntm_addr_kernel(
        const float* score, const float* __restrict__ w_prev,
        const float* __restrict__ gate, const float* __restrict__ sharp,
        const float* __restrict__ shift,
        float* w_tmp, float* __restrict__ bsum) {
    const int b   = blockIdx.x;
    const int tid = threadIdx.x;
    __shared__ float buf[N];
    __shared__ float red[256];

    // pass 1: load scores to LDS, row max
    float lmax = -3.4e38f;
    for (int i = tid; i < N; i += 256) {
        float v = score[(size_t)b * N + i];
        buf[i] = v;
        lmax = fmaxf(lmax, v);
    }
    red[tid] = lmax;
    __syncthreads();
    for (int s = 128; s > 0; s >>= 1) {
        if (tid < s) red[tid] = fmaxf(red[tid], red[tid + s]);
        __syncthreads();
    }
    const float vmax = red[0];
    __syncthreads();

    // pass 2: exp + sum
    float lsum = 0.f;
    for (int i = tid; i < N; i += 256) {
        float e = expf(buf[i] - vmax);
        buf[i] = e;
        lsum += e;
    }
    red[tid] = lsum;
    __syncthreads();
    for (int s = 128; s > 0; s >>= 1) {
        if (tid < s) red[tid] += red[tid + s];
        __syncthreads();
    }
    const float inv_sum = 1.0f / red[0];
    __syncthreads();

    // pass 3: w_g = g*w_c + (1-g)*w_prev  (in-place in LDS)
    const float g = gate[b];
    for (int i = tid; i < N; i += 256) {
        float wc = buf[i] * inv_sum;
        buf[i] = g * wc + (1.0f - g) * w_prev[(size_t)b * N + i];
    }
    __syncthreads();

    // pass 4: out[i] = sum_k shift[k]*w_g[(i-1+k) mod N], then ^sharp
    const float s0 = shift[b * SR + 0];
    const float s1 = shift[b * SR + 1];
    const float s2 = shift[b * SR + 2];
    const float sh = sharp[b];
    float lsum2 = 0.f;
    for (int i = tid; i < N; i += 256) {
        float wl = buf[(i + N - 1) & (N - 1)];
        float wc = buf[i];
        float wr = buf[(i + 1) & (N - 1)];
        float wt = powf(s0 * wl + s1 * wc + s2 * wr, sh);
        w_tmp[(size_t)b * N + i] = wt;
        lsum2 += wt;
    }
    red[tid] = lsum2;
    __syncthreads();
    for (int s = 128; s > 0; s >>= 1) {
        if (tid < s) red[tid] += red[tid + s];
        __syncthreads();
    }
    if (tid == 0) bsum[b] = red[0];
}

// K3b: reduce 256 batch sums -> total (+EPS), fixed order
__global__ __launch_bounds__(256) void ntm_total_kernel(
        const float* __restrict__ bsum, float* __restrict__ total) {
    __shared__ float red[256];
    const int tid = threadIdx.x;
    red[tid] = bsum[tid];   // B == 256 exactly
    __syncthreads();
    for (int s = 128; s > 0; s >>= 1) {
        if (tid < s) red[tid] += red[tid + s];
        __syncthreads();
    }
    if (tid == 0) *total = red[0] + EPSF;
}

// K4: global normalization
__global__ __launch_bounds__(256) void ntm_norm_kernel(
        float* __restrict__ w, const float* __restrict__ total) {
    const float inv = 1.0f / *total;
    size_t idx = (size_t)blockIdx.x * 256 + threadIdx.x;
    w[idx] *= inv;
}

// ---------------------------------------------------------------------------
extern "C" void kernel_launch(void* const* d_in, const int* in_sizes, int n_in,
                              void* d_out, int out_size, void* d_ws, size_t ws_size,
                              hipStream_t stream) {
    const float* ctrl   = (const float*)d_in[0];   // [B, INW]
    const float* memory = (const float*)d_in[1];   // [B, N, M]
    const float* w_prev = (const float*)d_in[2];   // [B, N]
    const float* fc_w   = (const float*)d_in[3];   // [NOUT, INW]
    const float* fc_b   = (const float*)d_in[4];   // [NOUT]
    float* out = (float*)d_out;                    // [B, N]

    float* wsf      = (float*)d_ws;
    float* key      = wsf;              // B*M
    float* key_norm = key + B * M;      // B
    float* stren    = key_norm + B;     // B
    float* gate     = stren + B;        // B
    float* sharp    = gate + B;         // B
    float* shift    = sharp + B;        // B*SR
    float* bsum     = shift + B * SR;   // B
    float* total    = bsum + B;         // 1

    ntm_fc_kernel<<<B, 128, 0, stream>>>(ctrl, fc_w, fc_b, key, key_norm,
                                         stren, gate, sharp, shift);

    dim3 g2(N / (16 * K2_WAVES), B);   // 64 x 256 blocks
    ntm_score_kernel<<<g2, 256, 0, stream>>>(memory, key, key_norm, stren, out);

    // scores live in d_out; K3 reads them into LDS, writes w_tmp back to d_out
    ntm_addr_kernel<<<B, 256, 0, stream>>>(out, w_prev, gate, sharp, shift,
                                           out, bsum);

    ntm_total_kernel<<<1, 256, 0, stream>>>(bsum, total);

    ntm_norm_kernel<<<(B * N) / 256, 256, 0, stream>>>(out, total);
}